// B_Splines_5866925326590
// MI455X (gfx1250) — compile-verified
//
#include <hip/hip_runtime.h>

// Cubic B-spline evaluation, sparse de Boor formulation.
// y[j] = sum_{k=0..3} Nb_j[k] * coefs[span_j - 3 + k]
// Final dot products executed on the matrix pipe via V_WMMA_F32_16X16X4_F32
// (full f32 precision), 16 points per WMMA, diagonal of D = y.

typedef __attribute__((ext_vector_type(2))) float v2f;
typedef __attribute__((ext_vector_type(8))) float v8f;

#define DEG 3
#define MAX_KNOTS 512
#define MAX_COEFS 512

// Largest s in [DEG, m-DEG-2] with t[s] <= x  (=> t[s] <= x < t[s+1] for valid x)
__device__ __forceinline__ int find_span(float x, const float* __restrict__ t, int m) {
  int lo = DEG;
  int hi = m - DEG - 2;
  while (lo < hi) {
    int mid = (lo + hi + 1) >> 1;
    if (t[mid] <= x) lo = mid; else hi = mid - 1;
  }
  return lo;
}

// NURBS-book A2.2: the DEG+1 nonzero basis functions at x for span s.
// Identical values (up to fp reassociation) to the reference Cox-de Boor recursion.
__device__ __forceinline__ void basis_funs(float x, int s, const float* __restrict__ t,
                                           float* __restrict__ Nb) {
  float left[DEG + 1], right[DEG + 1];
  Nb[0] = 1.0f;
#pragma unroll
  for (int j = 1; j <= DEG; ++j) {
    left[j]  = x - t[s + 1 - j];
    right[j] = t[s + j] - x;
    float saved = 0.0f;
#pragma unroll
    for (int r = 0; r < j; ++r) {
      float denom = right[r + 1] + left[j - r];   // t[s+r+1] - t[s+r+1-j], nonzero on valid spans
      float temp  = Nb[r] / denom;
      Nb[r] = saved + right[r + 1] * temp;
      saved = left[j - r] * temp;
    }
    Nb[j] = saved;
  }
}

__global__ void __launch_bounds__(256)
bspline_wmma_kernel(const float* __restrict__ x,
                    const float* __restrict__ knots,
                    const float* __restrict__ coefs,
                    float* __restrict__ out,
                    int N, int m, int n) {
  __shared__ float sk[MAX_KNOTS];
  __shared__ float sc[MAX_COEFS];
  for (int i = threadIdx.x; i < m; i += blockDim.x) sk[i] = knots[i];
  for (int i = threadIdx.x; i < n; i += blockDim.x) sc[i] = coefs[i];
  __syncthreads();

  const int lane          = threadIdx.x & 31;          // wave32
  const int waves_per_blk = blockDim.x >> 5;
  const int gwave         = blockIdx.x * waves_per_blk + (threadIdx.x >> 5);
  const int nwaves        = gridDim.x * waves_per_blk;

  const int ngroups = N >> 4;                          // 16 points per WMMA group
  const bool hi_half = (lane >= 16);

  for (int g = gwave; g < ngroups; g += nwaves) {      // wave-uniform loop: EXEC stays full
    const int base = g << 4;
    const int pt   = base + (lane & 15);               // lane pair (L, L+16) -> same point
    const float xv = x[pt];

    const int s = find_span(xv, sk, m);
    float Nb[DEG + 1];
    basis_funs(xv, s, sk, Nb);

    // A (16x4 f32): lanes 0-15 carry K=0,1 ; lanes 16-31 carry K=2,3 (same M = lane&15)
    v2f a, b;
    a[0] = hi_half ? Nb[2] : Nb[0];
    a[1] = hi_half ? Nb[3] : Nb[1];
    // B (4x16 f32): column j = coefs[span_j-3 .. span_j]; VGPR0 = rows K=0|K=2, VGPR1 = K=1|K=3
    const int ci = s - DEG + (hi_half ? 2 : 0);
    b[0] = sc[ci];
    b[1] = sc[ci + 1];

    v8f c = {};
    // D = A x B + 0 ; diag(D)[j] = y[base + j]
    c = __builtin_amdgcn_wmma_f32_16x16x4_f32(false, a, false, b, (short)0, c,
                                              false, false);

    // Diagonal extraction from the documented C/D layout:
    //   j in 0..7  -> lane j,     VGPR j
    //   j in 8..15 -> lane j+16,  VGPR j-8
    const int q = lane & 7;
    float y = c[0];
    y = (q == 1) ? c[1] : y;
    y = (q == 2) ? c[2] : y;
    y = (q == 3) ? c[3] : y;
    y = (q == 4) ? c[4] : y;
    y = (q == 5) ? c[5] : y;
    y = (q == 6) ? c[6] : y;
    y = (q == 7) ? c[7] : y;

    if (lane < 8)        out[base + lane]        = y;
    else if (lane >= 24) out[base + (lane - 16)] = y;
  }

  // Scalar tail for N % 16 (not taken for N = 500000, kept for generality).
  const int tid = blockIdx.x * blockDim.x + threadIdx.x;
  const int nth = gridDim.x * blockDim.x;
  for (int i = (ngroups << 4) + tid; i < N; i += nth) {
    const float xv = x[i];
    const int s = find_span(xv, sk, m);
    float Nb[DEG + 1];
    basis_funs(xv, s, sk, Nb);
    out[i] = Nb[0] * sc[s - 3] + Nb[1] * sc[s - 2] + Nb[2] * sc[s - 1] + Nb[3] * sc[s];
  }
}

extern "C" void kernel_launch(void* const* d_in, const int* in_sizes, int n_in,
                              void* d_out, int out_size, void* d_ws, size_t ws_size,
                              hipStream_t stream) {
  const float* x     = (const float*)d_in[0];
  const float* knots = (const float*)d_in[1];   // m = 260
  const float* coefs = (const float*)d_in[2];   // n = 256
  float* out = (float*)d_out;                   // N = 500000 floats

  const int N = in_sizes[0];
  const int m = in_sizes[1];
  const int n = in_sizes[2];

  const int threads       = 256;                // 8 wave32 per block
  const int waves_per_blk = threads / 32;
  const int ngroups       = N >> 4;
  int blocks = (ngroups + waves_per_blk - 1) / waves_per_blk;
  if (blocks < 1)    blocks = 1;
  if (blocks > 8192) blocks = 8192;             // grid-stride covers the rest

  bspline_wmma_kernel<<<blocks, threads, 0, stream>>>(x, knots, coefs, out, N, m, n);
}